// NewtonNet_7181185319449
// MI455X (gfx1250) — compile-verified
//
#include <hip/hip_runtime.h>
#include <cstddef>
#include <cstdint>

typedef float v2f __attribute__((ext_vector_type(2)));
typedef float v8f __attribute__((ext_vector_type(8)));

#define Bb 4
#define Aa 256
#define Nn 64
#define Ff 128
#define NBASIS 20
#define NLAYERS 3
#define BA (Bb*Aa)            /* 1024 atoms total   */
#define Ee (BA*Nn)            /* 65536 edges        */
#define CUT 5.0f
#define EPSV 1e-8f
#define PI_F 3.14159265358979323846f

#define OUT_INV 0
#define OUT_EQF_VEC (BA*Ff)                    /* 131072 */
#define OUT_EQF (OUT_EQF_VEC + BA*3)           /* 134144 */
#define OUT_EQDR (OUT_EQF + BA*3*Ff)           /* 527360 */
#define OUT_TOT (OUT_EQDR + BA*3*Ff)           /* 920576 */

// CDNA5 async global->LDS staging path (ASYNCcnt-tracked). Set to 0 to fall
// back to plain load+ds_store staging if the mnemonic is rejected.
#define ASYNC_LDS 1

__device__ __forceinline__ void async_g2l_b128(unsigned lds_off, const void* gaddr) {
#if ASYNC_LDS
  asm volatile("global_load_async_to_lds_b128 %0, %1, off"
               :: "v"(lds_off), "v"(gaddr) : "memory");
#endif
}
__device__ __forceinline__ void async_wait0() {
#if ASYNC_LDS
  asm volatile("s_wait_asynccnt 0x0" ::: "memory");
#endif
}

__device__ __forceinline__ v8f wmma_f32(v2f a, v2f b, v8f c) {
  // D = A(16x4, f32) * B(4x16, f32) + C(16x16, f32)
  return __builtin_amdgcn_wmma_f32_16x16x4_f32(false, a, false, b, (short)0, c, false, false);
}

__device__ __forceinline__ float silu(float x) {
  return x / (1.0f + expf(-x));
}

// ---------------------------------------------------------------------------
// Init: inv_node = emb[atomic_numbers]; zero eq_F / eq_f / eq_dr.
// ---------------------------------------------------------------------------
__global__ __launch_bounds__(256) void k_init_out(const int* __restrict__ an,
                                                  const float* __restrict__ emb,
                                                  float* __restrict__ out) {
  int idx = blockIdx.x * 256 + threadIdx.x;
  if (idx >= OUT_TOT) return;
  if (idx < BA * Ff) {
    int atom = idx >> 7, f = idx & 127;
    out[idx] = emb[an[atom] * Ff + f];
  } else {
    out[idx] = 0.0f;
  }
}

// ---------------------------------------------------------------------------
// Per-edge precompute: RBF basis (E x 20), poly cutoff (E), unit vectors (E x 3)
// ---------------------------------------------------------------------------
__global__ __launch_bounds__(256) void k_init_edges(const float* __restrict__ dist,
                                                    const float* __restrict__ dvec,
                                                    float* __restrict__ rbf,
                                                    float* __restrict__ cutb,
                                                    float* __restrict__ unitb) {
  int e = blockIdx.x * 256 + threadIdx.x;
  if (e >= Ee) return;
  float d = dist[e];
  float x = d * (1.0f / CUT);
  float x2 = x * x, x4 = x2 * x2, x6 = x4 * x2, x7 = x6 * x, x8 = x4 * x4;
  float f = 1.0f - 28.0f * x6 + 48.0f * x7 - 21.0f * x8;
  cutb[e] = (x < 1.0f) ? f : 0.0f;
  float inv = 1.0f / (d + EPSV);
  unitb[e * 3 + 0] = dvec[e * 3 + 0] * inv;
  unitb[e * 3 + 1] = dvec[e * 3 + 1] * inv;
  unitb[e * 3 + 2] = dvec[e * 3 + 2] * inv;
  float arg = PI_F * d * (1.0f / CUT);
  #pragma unroll
  for (int j = 0; j < NBASIS; ++j)
    rbf[e * NBASIS + j] = sinf((float)(j + 1) * arg) * inv;
}

// ---------------------------------------------------------------------------
// Fused 128->128->128 MLP: Y = silu(X @ W1^T + b1) @ W2^T + b2  (b1/b2 nullable)
// 32-row tile per block; wave w owns output cols [16w,16w+16) and computes two
// 16x16 C tiles (rows 0-15 / 16-31), reusing each B fragment for two WMMAs.
// X tile staged into LDS via async global->LDS b128 copies.
// ---------------------------------------------------------------------------
__global__ __launch_bounds__(256) void k_mlp128(const float* __restrict__ X,
                                                const float* __restrict__ W1,
                                                const float* __restrict__ b1,
                                                const float* __restrict__ W2,
                                                const float* __restrict__ b2,
                                                float* __restrict__ Y) {
  int row0 = blockIdx.x * 32;
  int tid = threadIdx.x;
  int w = tid >> 5, lane = tid & 31, r = lane & 15, half = lane >> 4;
  __shared__ __align__(16) float xs[32][132];
  __shared__ __align__(16) float hs[32][132];

#if ASYNC_LDS
  #pragma unroll
  for (int i = 0; i < 4; ++i) {
    int idx = tid + i * 256;              // 0..1023: 16B chunks of the 32x128 tile
    int rr = idx >> 5, c4 = (idx & 31) * 4;
    async_g2l_b128((unsigned)(uintptr_t)&xs[rr][c4],
                   X + (size_t)(row0 + rr) * Ff + c4);
  }
  async_wait0();
#else
  for (int i = tid; i < 32 * 128; i += 256) {
    int rr = i >> 7, cc = i & 127;
    xs[rr][cc] = X[(size_t)(row0 + rr) * Ff + cc];
  }
#endif
  __syncthreads();

  int col = (w << 4) + r;
  const float* w1r = W1 + (size_t)col * Ff;   // y[col] = sum_k x[k]*W1[col][k]
  v8f aL = {}, aH = {};
  #pragma unroll
  for (int k0 = 0; k0 < 128; k0 += 4) {
    int k = k0 + 2 * half;
    v2f b;   b.x = w1r[k];        b.y = w1r[k + 1];
    v2f alo; alo.x = xs[r][k];    alo.y = xs[r][k + 1];
    v2f ahi; ahi.x = xs[16 + r][k]; ahi.y = xs[16 + r][k + 1];
    aL = wmma_f32(alo, b, aL);
    aH = wmma_f32(ahi, b, aH);
  }
  float bias1 = b1 ? b1[col] : 0.0f;
  #pragma unroll
  for (int v = 0; v < 8; ++v) {
    hs[v + 8 * half][col]      = silu(aL[v] + bias1);
    hs[16 + v + 8 * half][col] = silu(aH[v] + bias1);
  }
  __syncthreads();

  const float* w2r = W2 + (size_t)col * Ff;
  v8f cL = {}, cH = {};
  #pragma unroll
  for (int k0 = 0; k0 < 128; k0 += 4) {
    int k = k0 + 2 * half;
    v2f b;   b.x = w2r[k];        b.y = w2r[k + 1];
    v2f alo; alo.x = hs[r][k];    alo.y = hs[r][k + 1];
    v2f ahi; ahi.x = hs[16 + r][k]; ahi.y = hs[16 + r][k + 1];
    cL = wmma_f32(alo, b, cL);
    cH = wmma_f32(ahi, b, cH);
  }
  float bias2 = b2 ? b2[col] : 0.0f;
  #pragma unroll
  for (int v = 0; v < 8; ++v) {
    Y[(size_t)(row0 + v + 8 * half) * Ff + col]      = cL[v] + bias2;
    Y[(size_t)(row0 + 16 + v + 8 * half) * Ff + col] = cH[v] + bias2;
  }
}

// ---------------------------------------------------------------------------
// inv_msg = ((rbf @ me_W^T + me_b) * cutoff) * msg_node[i] * msg_node[neighbor]
// 16 edges per block, K=20 via 5 WMMA steps. The per-edge eqc dot
// (eq_msg_F scalar) is fused here: xor-shuffle reduce over the 16-lane halves,
// then a tiny LDS reduction across the 8 waves.
// ---------------------------------------------------------------------------
__global__ __launch_bounds__(256) void k_edge_msg(const float* __restrict__ rbf,
                                                  const float* __restrict__ cutb,
                                                  const int* __restrict__ nbrs,
                                                  const float* __restrict__ msg_node,
                                                  const float* __restrict__ meW,
                                                  const float* __restrict__ meb,
                                                  const float* __restrict__ eqc,
                                                  float* __restrict__ inv_msg,
                                                  float* __restrict__ eqFb) {
  int e0 = blockIdx.x * 16;
  int tid = threadIdx.x;
  int w = tid >> 5, lane = tid & 31, r = lane & 15, half = lane >> 4;
  __shared__ float es[16][24];
  __shared__ float cut[16];
  __shared__ int arow[16];
  __shared__ int nrow[16];
  __shared__ float red[8][16];

  for (int i = tid; i < 16 * NBASIS; i += 256) {
    int rr = i / NBASIS, cc = i % NBASIS;
    es[rr][cc] = rbf[(size_t)(e0 + rr) * NBASIS + cc];
  }
  if (tid < 16) {
    int e = e0 + tid;
    cut[tid] = cutb[e];
    int atom = e >> 6;                        // edge -> atom (e/N)
    arow[tid] = atom;
    nrow[tid] = (atom >> 8) * Aa + nbrs[e];   // b*A + neighbor
  }
  __syncthreads();

  int col = (w << 4) + r;
  const float* wr = meW + (size_t)col * NBASIS;
  v8f acc = {};
  #pragma unroll
  for (int k0 = 0; k0 < NBASIS; k0 += 4) {
    v2f a, b;
    a.x = es[r][k0 + 2 * half];   a.y = es[r][k0 + 2 * half + 1];
    b.x = wr[k0 + 2 * half];      b.y = wr[k0 + 2 * half + 1];
    acc = wmma_f32(a, b, acc);
  }
  float bias = meb[col];
  float eqcv = eqc[col];
  float p[8];
  #pragma unroll
  for (int v = 0; v < 8; ++v) {
    int m = v + 8 * half;
    float val = (acc[v] + bias) * cut[m]
              * msg_node[(size_t)arow[m] * Ff + col]
              * msg_node[(size_t)nrow[m] * Ff + col];
    inv_msg[(size_t)(e0 + m) * Ff + col] = val;
    p[v] = val * eqcv;
  }
  // reduce each p[v] over the 16 lanes of this half (xor masks stay in-half)
  #pragma unroll
  for (int v = 0; v < 8; ++v) {
    #pragma unroll
    for (int m = 1; m < 16; m <<= 1) p[v] += __shfl_xor(p[v], m, 32);
  }
  if (r == 0) {
    #pragma unroll
    for (int v = 0; v < 8; ++v) red[w][v + 8 * half] = p[v];
  }
  __syncthreads();
  if (tid < 16) {
    float s = 0.0f;
    #pragma unroll
    for (int ww = 0; ww < 8; ++ww) s += red[ww][tid];
    eqFb[e0 + tid] = s;
  }
}

// ---------------------------------------------------------------------------
// Per-atom neighbor reduction for eq_F and eq_f:
//   s[n] = eqF[e]*mask[e];  eq_F += sum_n s*unit;  eq_upd_f[c,f] = sum_n feat*s*unit_c
// One block (128 threads) per atom.
// ---------------------------------------------------------------------------
__global__ __launch_bounds__(128) void k_accum_f(const float* __restrict__ feat,
                                                 const float* __restrict__ eqFb,
                                                 const float* __restrict__ nmask,
                                                 const float* __restrict__ unitb,
                                                 float* __restrict__ updf,
                                                 float* __restrict__ oeqF,
                                                 float* __restrict__ oeqf) {
  int atom = blockIdx.x;
  int tid = threadIdx.x;
  __shared__ float s[Nn], ux[Nn], uy[Nn], uz[Nn];
  if (tid < Nn) {
    int e = atom * Nn + tid;
    s[tid] = eqFb[e] * nmask[e];
    ux[tid] = unitb[e * 3 + 0];
    uy[tid] = unitb[e * 3 + 1];
    uz[tid] = unitb[e * 3 + 2];
  }
  __syncthreads();
  int f = tid;
  float a0 = 0.f, a1 = 0.f, a2 = 0.f;
  for (int n = 0; n < Nn; ++n) {
    float t = feat[(size_t)(atom * Nn + n) * Ff + f] * s[n];
    a0 += t * ux[n]; a1 += t * uy[n]; a2 += t * uz[n];
  }
  size_t base = (size_t)atom * 3 * Ff + f;
  updf[base + 0 * Ff] = a0;  oeqf[base + 0 * Ff] += a0;
  updf[base + 1 * Ff] = a1;  oeqf[base + 1 * Ff] += a1;
  updf[base + 2 * Ff] = a2;  oeqf[base + 2 * Ff] += a2;
  if (tid < 3) {
    float acc = 0.f;
    for (int n = 0; n < Nn; ++n)
      acc += s[n] * (tid == 0 ? ux[n] : (tid == 1 ? uy[n] : uz[n]));
    oeqF[atom * 3 + tid] += acc;
  }
}

// ---------------------------------------------------------------------------
// eq_upd_dr[c,f] = sum_n dr_edge[e,f] * eq_dr[neighbor,c,f] * mask[e]
// (reads the pre-update eq_dr; result written to scratch)
// ---------------------------------------------------------------------------
__global__ __launch_bounds__(128) void k_accum_dr(const float* __restrict__ dredge,
                                                  const int* __restrict__ nbrs,
                                                  const float* __restrict__ nmask,
                                                  const float* __restrict__ eqdr,
                                                  float* __restrict__ upddr) {
  int atom = blockIdx.x;
  int tid = threadIdx.x;
  __shared__ int nb[Nn];
  __shared__ float mk[Nn];
  if (tid < Nn) {
    int e = atom * Nn + tid;
    nb[tid] = (atom >> 8) * Aa + nbrs[e];
    mk[tid] = nmask[e];
  }
  __syncthreads();
  int f = tid;
  float a0 = 0.f, a1 = 0.f, a2 = 0.f;
  for (int n = 0; n < Nn; ++n) {
    float dv = dredge[(size_t)(atom * Nn + n) * Ff + f] * mk[n];
    size_t nbase = (size_t)nb[n] * 3 * Ff + f;
    a0 += dv * eqdr[nbase + 0 * Ff];
    a1 += dv * eqdr[nbase + 1 * Ff];
    a2 += dv * eqdr[nbase + 2 * Ff];
  }
  size_t base = (size_t)atom * 3 * Ff + f;
  upddr[base + 0 * Ff] = a0;
  upddr[base + 1 * Ff] = a1;
  upddr[base + 2 * Ff] = a2;
}

// eq_dr += eq_upd_dr + esu(inv_node)[f] * eq_upd_f[c,f]
__global__ __launch_bounds__(256) void k_update_dr(const float* __restrict__ upddr,
                                                   const float* __restrict__ esu,
                                                   const float* __restrict__ updf,
                                                   float* __restrict__ eqdr) {
  int idx = blockIdx.x * 256 + threadIdx.x;
  if (idx >= BA * 3 * Ff) return;
  int atom = idx / (3 * Ff);
  int f = idx & (Ff - 1);
  eqdr[idx] += upddr[idx] + esu[atom * Ff + f] * updf[idx];
}

// inv_node += -isu(inv_node) * sum_c eq_f[c,f]*eq_dr[c,f]
__global__ __launch_bounds__(256) void k_update_inv(const float* __restrict__ oeqf,
                                                    const float* __restrict__ eqdr,
                                                    const float* __restrict__ isu,
                                                    float* __restrict__ inv) {
  int idx = blockIdx.x * 256 + threadIdx.x;
  if (idx >= BA * Ff) return;
  int atom = idx >> 7, f = idx & 127;
  size_t base = (size_t)atom * 3 * Ff + f;
  float dot = oeqf[base] * eqdr[base]
            + oeqf[base + Ff] * eqdr[base + Ff]
            + oeqf[base + 2 * Ff] * eqdr[base + 2 * Ff];
  inv[idx] -= isu[idx] * dot;
}

// ---------------------------------------------------------------------------
extern "C" void kernel_launch(void* const* d_in, const int* in_sizes, int n_in,
                              void* d_out, int out_size, void* d_ws, size_t ws_size,
                              hipStream_t stream) {
  const int*   an    = (const int*)d_in[0];
  // d_in[1] = positions (unused by reference forward)
  const int*   nbrs  = (const int*)d_in[2];
  const float* nmask = (const float*)d_in[3];
  const float* dist  = (const float*)d_in[4];
  const float* dvec  = (const float*)d_in[5];
  const float* emb   = (const float*)d_in[6];
  const float* me_W  = (const float*)d_in[7];
  const float* me_b  = (const float*)d_in[8];
  const float* mn_W1 = (const float*)d_in[9];
  const float* mn_b1 = (const float*)d_in[10];
  const float* mn_W2 = (const float*)d_in[11];
  const float* mn_b2 = (const float*)d_in[12];
  const float* eqc_W = (const float*)d_in[13];
  const float* eqf_W1 = (const float*)d_in[14];
  const float* eqf_b1 = (const float*)d_in[15];
  const float* eqf_W2 = (const float*)d_in[16];
  const float* eqf_b2 = (const float*)d_in[17];
  const float* esu_W1 = (const float*)d_in[18];
  const float* esu_b1 = (const float*)d_in[19];
  const float* esu_W2 = (const float*)d_in[20];
  const float* esu_b2 = (const float*)d_in[21];
  const float* eme_W1 = (const float*)d_in[22];
  const float* eme_W2 = (const float*)d_in[23];
  const float* isu_W1 = (const float*)d_in[24];
  const float* isu_b1 = (const float*)d_in[25];
  const float* isu_W2 = (const float*)d_in[26];
  const float* isu_b2 = (const float*)d_in[27];

  float* out   = (float*)d_out;
  float* o_inv  = out + OUT_INV;
  float* o_eqF  = out + OUT_EQF_VEC;
  float* o_eqf  = out + OUT_EQF;
  float* o_eqdr = out + OUT_EQDR;

  // workspace layout (floats)
  float* ws = (float*)d_ws;
  float* w_rbf   = ws;                              // E*20
  float* w_cut   = w_rbf   + (size_t)Ee * NBASIS;   // E
  float* w_unit  = w_cut   + Ee;                    // E*3
  float* w_eqF   = w_unit  + (size_t)Ee * 3;        // E
  float* w_msgn  = w_eqF   + Ee;                    // BA*F
  float* w_esu   = w_msgn  + BA * Ff;               // BA*F
  float* w_isu   = w_esu   + BA * Ff;               // BA*F
  float* w_updf  = w_isu   + BA * Ff;               // BA*3*F
  float* w_upddr = w_updf  + BA * 3 * Ff;           // BA*3*F
  float* w_imsg  = w_upddr + BA * 3 * Ff;           // E*F
  float* w_efeat = w_imsg  + (size_t)Ee * Ff;       // E*F

  k_init_out<<<(OUT_TOT + 255) / 256, 256, 0, stream>>>(an, emb, out);
  k_init_edges<<<Ee / 256, 256, 0, stream>>>(dist, dvec, w_rbf, w_cut, w_unit);

  for (int l = 0; l < NLAYERS; ++l) {
    const size_t oW = (size_t)l * Ff * Ff, ob = (size_t)l * Ff;
    // node MLPs on inv_node (pre-update), all done up front
    k_mlp128<<<BA / 32, 256, 0, stream>>>(o_inv, mn_W1 + oW, mn_b1 + ob,
                                          mn_W2 + oW, mn_b2 + ob, w_msgn);
    k_mlp128<<<BA / 32, 256, 0, stream>>>(o_inv, esu_W1 + oW, esu_b1 + ob,
                                          esu_W2 + oW, esu_b2 + ob, w_esu);
    k_mlp128<<<BA / 32, 256, 0, stream>>>(o_inv, isu_W1 + oW, isu_b1 + ob,
                                          isu_W2 + oW, isu_b2 + ob, w_isu);
    // inv_msg on edges (RBF GEMM fused with cutoff, node gathers, eqc dot)
    k_edge_msg<<<Ee / 16, 256, 0, stream>>>(w_rbf, w_cut, nbrs, w_msgn,
                                            me_W + (size_t)l * Ff * NBASIS,
                                            me_b + ob, eqc_W + ob, w_imsg, w_eqF);
    // eq feature MLP on edges + neighbor reduction
    k_mlp128<<<Ee / 32, 256, 0, stream>>>(w_imsg, eqf_W1 + oW, eqf_b1 + ob,
                                          eqf_W2 + oW, eqf_b2 + ob, w_efeat);
    k_accum_f<<<BA, 128, 0, stream>>>(w_efeat, w_eqF, nmask, w_unit,
                                      w_updf, o_eqF, o_eqf);
    // dr edge MLP (no biases) + gather-reduce against old eq_dr
    k_mlp128<<<Ee / 32, 256, 0, stream>>>(w_imsg, eme_W1 + oW, nullptr,
                                          eme_W2 + oW, nullptr, w_efeat);
    k_accum_dr<<<BA, 128, 0, stream>>>(w_efeat, nbrs, nmask, o_eqdr, w_upddr);
    k_update_dr<<<(BA * 3 * Ff) / 256, 256, 0, stream>>>(w_upddr, w_esu, w_updf, o_eqdr);
    // invariant node update
    k_update_inv<<<(BA * Ff) / 256, 256, 0, stream>>>(o_eqf, o_eqdr, w_isu, o_inv);
  }
  (void)in_sizes; (void)n_in; (void)out_size; (void)ws_size;
}